// Model_21818433863799
// MI455X (gfx1250) — compile-verified
//
#include <hip/hip_runtime.h>

typedef __attribute__((ext_vector_type(2))) float v2f;
typedef __attribute__((ext_vector_type(8))) float v8f;

#define NU 100000
#define NM 20000
#define NE 1000000
#define NL 500000

// ---------------- zero scratch ----------------
__global__ void zero_kernel(float* __restrict__ p, int n) {
    int i = blockIdx.x * blockDim.x + threadIdx.x;
    const int stride = gridDim.x * blockDim.x;
    for (; i < n; i += stride) p[i] = 0.0f;
}

// ---------------- edge scatter-add (one wave per edge iteration) ----------------
// lane l accumulates elements [2l, 2l+1] of the 64-wide source row into agg[dst].
__global__ void scatter_mean_kernel(const float* __restrict__ xsrc,
                                    const int* __restrict__ src,
                                    const int* __restrict__ dst,
                                    float* __restrict__ agg,
                                    float* __restrict__ cnt) {
    const int lane = threadIdx.x & 31;
    int w = (blockIdx.x * blockDim.x + threadIdx.x) >> 5;
    const int nw = (gridDim.x * blockDim.x) >> 5;
    for (int e = w; e < NE; e += nw) {
        const int s = src[e];
        const int d = dst[e];
        const float2 v = *(const float2*)(xsrc + s * 64 + lane * 2);
        float* o = agg + d * 64 + lane * 2;
        atomicAdd(o + 0, v.x);
        atomicAdd(o + 1, v.y);
        if (lane == 0) atomicAdd(cnt + d, 1.0f);
    }
}

// ---------------- SAGE layer GEMM: out = (agg/max(cnt,1)) @ Wl + xdst @ Wr + b ----------------
// One wave per 16-row tile; 8 waves per block share Wl/Wr/bias in LDS.
// Uses V_WMMA_F32_16X16X4_F32 (fp32 in/out, 16x16 tiles, K stepped by 4).
__global__ void __launch_bounds__(256) sage_gemm_kernel(
        const float* __restrict__ agg, const float* __restrict__ cnt,
        const float* __restrict__ xdst,
        const float* __restrict__ Wl, const float* __restrict__ Wr,
        const float* __restrict__ bias,
        float* __restrict__ out, int nrows, int relu) {
    __shared__ float sWl[64 * 64];
    __shared__ float sWr[64 * 64];
    __shared__ float sB[64];

    const int t = threadIdx.x;
    for (int i = t; i < 64 * 64; i += 256) {
        sWl[i] = Wl[i];
        sWr[i] = Wr[i];
    }
    if (t < 64) sB[t] = bias[t];
    __syncthreads();

    const int wave = t >> 5;
    const int lane = t & 31;
    const int tile = blockIdx.x * 8 + wave;
    if (tile * 16 >= nrows) return;   // wave-uniform: EXEC stays all-1s for WMMA waves

    const int half = lane >> 4;       // 0: K even pair / rows 0-7 ; 1: K odd pair / rows 8-15
    const int m16  = lane & 15;
    const int row  = tile * 16 + m16; // A-operand row handled by this lane
    const float inv = 1.0f / fmaxf(cnt[row], 1.0f);

    v8f acc[4] = {{}, {}, {}, {}};

    // GEMM 1: mean @ Wl   (scale A on load by 1/count)
    for (int kk = 0; kk < 16; ++kk) {
        const int kbase = kk * 4 + 2 * half;
        const float2 av = *(const float2*)(agg + row * 64 + kbase);
        v2f a; a.x = av.x * inv; a.y = av.y * inv;
#pragma unroll
        for (int nt = 0; nt < 4; ++nt) {
            const int n = nt * 16 + m16;
            v2f bm;
            bm.x = sWl[kbase * 64 + n];
            bm.y = sWl[(kbase + 1) * 64 + n];
            acc[nt] = __builtin_amdgcn_wmma_f32_16x16x4_f32(
                false, a, false, bm, (short)0, acc[nt], false, false);
        }
    }

    // GEMM 2: xdst @ Wr
    for (int kk = 0; kk < 16; ++kk) {
        const int kbase = kk * 4 + 2 * half;
        const float2 av = *(const float2*)(xdst + row * 64 + kbase);
        v2f a; a.x = av.x; a.y = av.y;
#pragma unroll
        for (int nt = 0; nt < 4; ++nt) {
            const int n = nt * 16 + m16;
            v2f bm;
            bm.x = sWr[kbase * 64 + n];
            bm.y = sWr[(kbase + 1) * 64 + n];
            acc[nt] = __builtin_amdgcn_wmma_f32_16x16x4_f32(
                false, a, false, bm, (short)0, acc[nt], false, false);
        }
    }

    // store: c[j] of lane l is row (j + 8*half), col (nt*16 + m16)
#pragma unroll
    for (int nt = 0; nt < 4; ++nt) {
        const int c = nt * 16 + m16;
        const float bv = sB[c];
#pragma unroll
        for (int j = 0; j < 8; ++j) {
            const int r = tile * 16 + j + 8 * half;
            float v = acc[nt][j] + bv;
            if (relu) v = fmaxf(v, 0.0f);
            out[r * 64 + c] = v;
        }
    }
}

// ---------------- decoder: out[i] = dot(xu[es[i]], xm[ed[i]]) ----------------
// Half-wave (16 lanes x float4) per pair -> one full coalesced 256B row each.
__global__ void decode_kernel(const float* __restrict__ xu, const float* __restrict__ xm,
                              const int* __restrict__ es, const int* __restrict__ ed,
                              float* __restrict__ out) {
    const int t = blockIdx.x * blockDim.x + threadIdx.x;
    const int pair = t >> 4;
    const int l16 = t & 15;
    if (pair >= NL) return;
    const int u = es[pair];
    const int m = ed[pair];
    const float4 a = *(const float4*)(xu + (size_t)u * 64 + l16 * 4);
    const float4 b = *(const float4*)(xm + (size_t)m * 64 + l16 * 4);
    float s = a.x * b.x + a.y * b.y + a.z * b.z + a.w * b.w;
#pragma unroll
    for (int off = 8; off >= 1; off >>= 1) s += __shfl_xor(s, off, 16);
    if (l16 == 0) out[pair] = s;
}

extern "C" void kernel_launch(void* const* d_in, const int* in_sizes, int n_in,
                              void* d_out, int out_size, void* d_ws, size_t ws_size,
                              hipStream_t stream) {
    const float* x_user  = (const float*)d_in[0];
    const float* x_movie = (const float*)d_in[1];
    const float* wl      = (const float*)d_in[2];   // [3,2,64,64]
    const float* wr      = (const float*)d_in[3];   // [3,2,64,64]
    const float* bb      = (const float*)d_in[4];   // [3,2,64]
    const int* src_um    = (const int*)d_in[5];
    const int* dst_um    = (const int*)d_in[6];
    const int* src_mu    = (const int*)d_in[7];
    const int* dst_mu    = (const int*)d_in[8];
    const int* ell_src   = (const int*)d_in[9];
    const int* ell_dst   = (const int*)d_in[10];
    float* out = (float*)d_out;

    char* ws = (char*)d_ws;
    // workspace layout (bytes)
    float* xu_b[2] = { (float*)(ws + 0),          (float*)(ws + 25600000) };
    float* xm_b[2] = { (float*)(ws + 51200000),   (float*)(ws + 56320000) };
    float* agg_u   = (float*)(ws + 61440000);     // 25,600,000 B
    float* agg_m   = (float*)(ws + 87040000);     //  5,120,000 B
    float* cnt_u   = (float*)(ws + 92160000);     //    400,000 B
    float* cnt_m   = (float*)(ws + 92560000);     //     80,000 B
    // agg_u..cnt_m are contiguous: 31,200,000 B = 7,800,000 floats

    const float* xu_cur = x_user;
    const float* xm_cur = x_movie;

    for (int layer = 0; layer < 3; ++layer) {
        zero_kernel<<<4096, 256, 0, stream>>>(agg_u, 7800000);

        // um edges: user -> movie (dst = movie); mu edges: movie -> user (dst = user)
        scatter_mean_kernel<<<4096, 256, 0, stream>>>(xu_cur, src_um, dst_um, agg_m, cnt_m);
        scatter_mean_kernel<<<4096, 256, 0, stream>>>(xm_cur, src_mu, dst_mu, agg_u, cnt_u);

        const int relu = (layer < 2) ? 1 : 0;
        const float* Wl0 = wl + (size_t)(layer * 2 + 0) * 4096;
        const float* Wr0 = wr + (size_t)(layer * 2 + 0) * 4096;
        const float* B0  = bb + (size_t)(layer * 2 + 0) * 64;
        const float* Wl1 = wl + (size_t)(layer * 2 + 1) * 4096;
        const float* Wr1 = wr + (size_t)(layer * 2 + 1) * 4096;
        const float* B1  = bb + (size_t)(layer * 2 + 1) * 64;

        float* xm_next = xm_b[layer & 1];
        float* xu_next = xu_b[layer & 1];

        // movies: 20000/16 = 1250 tiles, 8 tiles/block
        sage_gemm_kernel<<<(1250 + 7) / 8, 256, 0, stream>>>(
            agg_m, cnt_m, xm_cur, Wl0, Wr0, B0, xm_next, NM, relu);
        // users: 100000/16 = 6250 tiles
        sage_gemm_kernel<<<(6250 + 7) / 8, 256, 0, stream>>>(
            agg_u, cnt_u, xu_cur, Wl1, Wr1, B1, xu_next, NU, relu);

        xu_cur = xu_next;
        xm_cur = xm_next;
    }

    // 500000 pairs * 16 lanes = 8,000,000 threads = 31250 blocks of 256
    decode_kernel<<<31250, 256, 0, stream>>>(xu_cur, xm_cur, ell_src, ell_dst, out);
}